// PatchNCELoss_34059090657625
// MI455X (gfx1250) — compile-verified
//
#include <hip/hip_runtime.h>
#include <math.h>

typedef __attribute__((ext_vector_type(2))) float v2f;
typedef __attribute__((ext_vector_type(8))) float v8f;

#define N_ROWS 8192
#define DIM 128
#define INV_TAU 50.0f          // 1 / 0.02
#define LDS_STRIDE 132         // 128 + 4 pad -> conflict-free ds_load_b64

// ---------------------------------------------------------------------------
// Kernel 1: row L2-normalize ts and sq into workspace; zero the accumulator.
// One wave per row (32 lanes x float4 = 128 elements).
// ---------------------------------------------------------------------------
__global__ __launch_bounds__(256) void nce_normalize(const float* __restrict__ ts,
                                                     const float* __restrict__ sq,
                                                     float* __restrict__ tsn,
                                                     float* __restrict__ sqn,
                                                     float* __restrict__ accum) {
  if (blockIdx.x == 0 && threadIdx.x == 0) *accum = 0.0f;
  const int wave = threadIdx.x >> 5;
  const int lane = threadIdx.x & 31;
  const int row  = blockIdx.x * 8 + wave;          // 0..16383 (ts then sq)
  const float* src;
  float* dst;
  if (row < N_ROWS) { src = ts + (size_t)row * DIM;            dst = tsn + (size_t)row * DIM; }
  else              { src = sq + (size_t)(row - N_ROWS) * DIM; dst = sqn + (size_t)(row - N_ROWS) * DIM; }

  float4 v = *(const float4*)(src + lane * 4);
  float ss = v.x * v.x + v.y * v.y + v.z * v.z + v.w * v.w;
#pragma unroll
  for (int off = 16; off >= 1; off >>= 1) ss += __shfl_xor(ss, off, 32);
  const float s = 1.0f / fmaxf(sqrtf(ss), 1e-12f);   // F.normalize eps
  float4 o;
  o.x = v.x * s; o.y = v.y * s; o.z = v.z * s; o.w = v.w * s;
  *(float4*)(dst + lane * 4) = o;
}

// ---------------------------------------------------------------------------
// Kernel 2: fused logits GEMM (fp32 WMMA 16x16x4) + online row LSE + diagonal.
// 4 waves/block, each wave owns a 16-row M tile; block streams all 8192
// columns in 16-wide tiles staged through LDS.
// ---------------------------------------------------------------------------
__global__ __launch_bounds__(128) void nce_main(const float* __restrict__ tsn,
                                                const float* __restrict__ sqn,
                                                const int* __restrict__ pmask,
                                                float* __restrict__ accum) {
  __shared__ __attribute__((aligned(16))) float sB[16 * LDS_STRIDE];
  __shared__ float sDiag[4][16];

  const int tid   = threadIdx.x;
  const int wave  = tid >> 5;
  const int lane  = tid & 31;
  const int row0  = blockIdx.x * 64 + wave * 16;

  const int m     = lane & 15;          // A: row; B: column
  const int khalf = (lane >> 4) * 2;    // K sub-offset within a K=4 step

  // Load A fragments once: 32 K-steps x v2f per lane (row m of the M tile).
  v2f A[32];
#pragma unroll
  for (int i = 0; i < 32; ++i)
    A[i] = *(const v2f*)(tsn + (size_t)(row0 + m) * DIM + i * 4 + khalf);

  // Which accumulator slot holds this lane's diagonal element (C layout):
  // lanes 0..7 -> rows 0..7 (slot = lane); lanes 24..31 -> rows 8..15 (slot = lane-24).
  int selIdx = -1;
  if (lane < 8)        selIdx = lane;
  else if (lane >= 24) selIdx = lane - 24;

  float rmax[8], rsum[8];
#pragma unroll
  for (int j = 0; j < 8; ++j) { rmax[j] = -INFINITY; rsum[j] = 0.0f; }
  float dg = 0.0f;

  const int r16   = tid >> 3;           // staging: row of B tile (0..15)
  const int cbase = (tid & 7) * 16;     // staging: column base

  for (int t = 0; t < N_ROWS / 16; ++t) {
    const int col0 = t * 16;

    // Cooperative stage of B tile (16 rows x 128 cols) into padded LDS.
#pragma unroll
    for (int q = 0; q < 4; ++q) {
      const int c = cbase + q * 4;
      float4 b4 = *(const float4*)(sqn + (size_t)(col0 + r16) * DIM + c);
      *(float4*)(&sB[r16 * LDS_STRIDE + c]) = b4;
    }
    if (t + 1 < N_ROWS / 16)
      __builtin_prefetch(sqn + (size_t)(col0 + 16 + r16) * DIM + cbase, 0, 0);
    __syncthreads();

    // 16x16 logits tile via 32 chained fp32 WMMAs over K=128.
    v8f c = {0.f, 0.f, 0.f, 0.f, 0.f, 0.f, 0.f, 0.f};
    const float* bptr = &sB[m * LDS_STRIDE + khalf];
#pragma unroll
    for (int i = 0; i < 32; ++i) {
      v2f b = *(const v2f*)(bptr + i * 4);
      c = __builtin_amdgcn_wmma_f32_16x16x4_f32(false, A[i], false, b,
                                                (short)0, c, false, false);
    }

    // Online softmax update (one v_exp_f32 per logit) + diagonal capture.
    const bool diagTile = (col0 == row0);
#pragma unroll
    for (int j = 0; j < 8; ++j) {
      const float v = c[j] * INV_TAU;
      if (diagTile && j == selIdx) dg = v;
      const float old = rmax[j];
      const float nm  = fmaxf(old, v);
      const float e   = __expf(fminf(old, v) - nm);   // exp(0-or-negative diff)
      rsum[j] = (v > old) ? (rsum[j] * e + 1.0f) : (rsum[j] + e);
      rmax[j] = nm;
    }
    __syncthreads();
  }

  // Park diagonal logits (per-wave region, same-wave readers).
  if (selIdx >= 0) {
    const int rloc = (lane < 8) ? lane : (lane - 16);   // 0..7 or 8..15
    sDiag[wave][rloc] = dg;
  }

  // Merge (max,sum) across the 16 lanes that share each row group.
#pragma unroll
  for (int j = 0; j < 8; ++j) {
#pragma unroll
    for (int off = 1; off < 16; off <<= 1) {
      const float m2 = __shfl_xor(rmax[j], off, 32);
      const float s2 = __shfl_xor(rsum[j], off, 32);
      const float nm = fmaxf(rmax[j], m2);
      rsum[j] = rsum[j] * __expf(rmax[j] - nm) + s2 * __expf(m2 - nm);
      rmax[j] = nm;
    }
  }
  __syncthreads();

  // Lane 0 -> rows row0+0..7, lane 16 -> rows row0+8..15.
  if ((lane & 15) == 0) {
    const int base = (lane >> 4) * 8;
    float part = 0.0f;
#pragma unroll
    for (int j = 0; j < 8; ++j) {
      const int r    = row0 + base + j;
      const float lse = rmax[j] + __logf(rsum[j]);
      const float d   = sDiag[wave][base + j];
      part += (float)pmask[r] * (d - lse);
    }
    atomicAdd(accum, part);
  }
}

// ---------------------------------------------------------------------------
// Kernel 3: patch_sum reduction + final scalar.
// ---------------------------------------------------------------------------
__global__ __launch_bounds__(256) void nce_finalize(const int* __restrict__ pmask,
                                                    const float* __restrict__ accum,
                                                    float* __restrict__ out) {
  __shared__ float red[256];
  float s = 0.0f;
  for (int i = threadIdx.x; i < N_ROWS; i += 256) s += (float)pmask[i];
  red[threadIdx.x] = s;
  __syncthreads();
  for (int w = 128; w > 0; w >>= 1) {
    if (threadIdx.x < w) red[threadIdx.x] += red[threadIdx.x + w];
    __syncthreads();
  }
  if (threadIdx.x == 0) out[0] = -accum[0] / (red[0] + 1e-6f);
}

// ---------------------------------------------------------------------------
extern "C" void kernel_launch(void* const* d_in, const int* in_sizes, int n_in,
                              void* d_out, int out_size, void* d_ws, size_t ws_size,
                              hipStream_t stream) {
  const float* ts = (const float*)d_in[0];
  const float* sq = (const float*)d_in[1];
  // d_in[2] = omega (unused by the forward pass)
  const int* pm = (const int*)d_in[3];

  float* tsn   = (float*)d_ws;
  float* sqn   = tsn + (size_t)N_ROWS * DIM;
  float* accum = sqn + (size_t)N_ROWS * DIM;

  nce_normalize<<<2048, 256, 0, stream>>>(ts, sq, tsn, sqn, accum);
  nce_main<<<128, 128, 0, stream>>>(tsn, sqn, pm, accum);
  nce_finalize<<<1, 256, 0, stream>>>(pm, accum, (float*)d_out);
}